// MultiLinear_48498770706571
// MI455X (gfx1250) — compile-verified
//
#include <hip/hip_runtime.h>

typedef __attribute__((ext_vector_type(2))) float v2f;
typedef __attribute__((ext_vector_type(8))) float v8f;
typedef __attribute__((ext_vector_type(4))) int   v4i;

#define BDIM 4096
#define GDIM 16
#define DIN  512
#define DOUT 512

#define BM 64
#define BN 128
#define BK 32
#define LSTR 36            // BK + 4 pad: 36*r mod 64 covers all multiples of 4 -> conflict-free b64 frag reads
#define NTILES (DIN / BK)  // 16

#if __has_builtin(__builtin_amdgcn_global_load_async_to_lds_b128)
#define USE_ASYNC 1
typedef __attribute__((address_space(1))) v4i* gas1_t;   // global source (b128 granule)
typedef __attribute__((address_space(3))) v4i* las3_t;   // LDS destination (b128 granule)
#else
#define USE_ASYNC 0
#endif

__global__ __launch_bounds__(256)
void multilinear_wmma_f32(const float* __restrict__ x,
                          const float* __restrict__ w,
                          const float* __restrict__ bias,
                          float* __restrict__ y)
{
    __shared__ float lsx[2][BM * LSTR];   // 2 x  9,216 B
    __shared__ float lsw[2][BN * LSTR];   // 2 x 18,432 B  (55 KB total of 320 KB/WGP)

    const int g     = blockIdx.z;
    const int bBase = blockIdx.y * BM;
    const int oBase = blockIdx.x * BN;

    const int tid  = threadIdx.x;
    const int wid  = tid >> 5;
    const int lane = tid & 31;
    const int lr   = lane & 15;   // row/col within 16-wide fragment
    const int lh   = lane >> 4;   // half-wave: K offset +2 on input frags, M offset +8 on output

    const int waveM = wid >> 2;   // 0..1 : 32-row slab of the 64-row block tile
    const int waveN = wid & 3;    // 0..3 : 32-col slab of the 128-col block tile

    // --- tile staging coordinates (256 threads, one b128 per row-octet) ---
    const int lrow = tid >> 3;        // 0..31
    const int lcol = (tid & 7) << 2;  // 0,4,...,28

    const float* xrow = x + ((size_t)(bBase + lrow) * GDIM + g) * DIN + lcol;   // advance by 32*G*DIN per pass
    const float* wrow = w + ((size_t)g * DOUT + oBase + lrow) * DIN + lcol;     // advance by 32*DIN per pass

#if USE_ASYNC
    // Async DMA global -> LDS (per-lane b128), tracked by ASYNCcnt. 6 instructions per tile.
    auto issue_tile = [&](int buf, int k0) {
#pragma unroll
        for (int p = 0; p < 2; ++p)
            __builtin_amdgcn_global_load_async_to_lds_b128(
                (gas1_t)(xrow + (size_t)p * 32 * GDIM * DIN + k0),
                (las3_t)&lsx[buf][(p * 32 + lrow) * LSTR + lcol], 0, 0);
#pragma unroll
        for (int p = 0; p < 4; ++p)
            __builtin_amdgcn_global_load_async_to_lds_b128(
                (gas1_t)(wrow + (size_t)p * 32 * DIN + k0),
                (las3_t)&lsw[buf][(p * 32 + lrow) * LSTR + lcol], 0, 0);
    };
#else
    // Fallback: synchronous regs -> LDS staging (no cross-iteration live registers).
    auto issue_tile = [&](int buf, int k0) {
#pragma unroll
        for (int p = 0; p < 2; ++p) {
            float4 t = *(const float4*)(xrow + (size_t)p * 32 * GDIM * DIN + k0);
            *(float4*)(&lsx[buf][(p * 32 + lrow) * LSTR + lcol]) = t;
        }
#pragma unroll
        for (int p = 0; p < 4; ++p) {
            float4 t = *(const float4*)(wrow + (size_t)p * 32 * DIN + k0);
            *(float4*)(&lsw[buf][(p * 32 + lrow) * LSTR + lcol]) = t;
        }
    };
#endif

    // --- accumulators, bias folded in (bias depends only on the output column = lane's N) ---
    v8f acc[2][2];
#pragma unroll
    for (int ni = 0; ni < 2; ++ni) {
        const float bv = bias[g * DOUT + oBase + waveN * 32 + ni * 16 + lr];
#pragma unroll
        for (int mi = 0; mi < 2; ++mi)
#pragma unroll
            for (int e = 0; e < 8; ++e)
                acc[mi][ni][e] = bv;
    }

    // fragment offsets in a tile buffer: lane reads (row = lr, cols kk..kk+1, half-wave +2)
    const int axOff = (waveM * 32 + lr) * LSTR + 2 * lh;
    const int bwOff = (waveN * 32 + lr) * LSTR + 2 * lh;

    auto compute_tile = [&](int buf) {
        const float* ax0 = &lsx[buf][axOff];
        const float* ax1 = ax0 + 16 * LSTR;
        const float* bw0 = &lsw[buf][bwOff];
        const float* bw1 = bw0 + 16 * LSTR;
#pragma unroll
        for (int kk = 0; kk < BK; kk += 4) {
            v2f a0 = *(const v2f*)(ax0 + kk);
            v2f a1 = *(const v2f*)(ax1 + kk);
            v2f b0 = *(const v2f*)(bw0 + kk);
            v2f b1 = *(const v2f*)(bw1 + kk);
            acc[0][0] = __builtin_amdgcn_wmma_f32_16x16x4_f32(false, a0, false, b0,
                                                              (short)0, acc[0][0], false, false);
            acc[0][1] = __builtin_amdgcn_wmma_f32_16x16x4_f32(false, a0, false, b1,
                                                              (short)0, acc[0][1], false, false);
            acc[1][0] = __builtin_amdgcn_wmma_f32_16x16x4_f32(false, a1, false, b0,
                                                              (short)0, acc[1][0], false, false);
            acc[1][1] = __builtin_amdgcn_wmma_f32_16x16x4_f32(false, a1, false, b1,
                                                              (short)0, acc[1][1], false, false);
        }
    };

#if USE_ASYNC
    issue_tile(0, 0);  // prologue DMA
#pragma unroll 1
    for (int t = 0; t < NTILES; ++t) {
        const int cur = t & 1;
        if (t + 1 < NTILES) {
            issue_tile(cur ^ 1, (t + 1) * BK);                 // next tile's DMA in flight
            asm volatile("s_wait_asynccnt 0x6" ::: "memory");  // tile t done (in-order), 6 newest may remain
        } else {
            asm volatile("s_wait_asynccnt 0x0" ::: "memory");
        }
        __syncthreads();       // tile t visible to all waves
        compute_tile(cur);     // 32 WMMAs
        __syncthreads();       // all waves done reading buf[cur] before it is re-targeted
    }
#else
#pragma unroll 1
    for (int t = 0; t < NTILES; ++t) {
        issue_tile(0, t * BK);
        __syncthreads();
        compute_tile(0);
        __syncthreads();
    }
#endif

    // --- epilogue: C/D layout = VGPR j -> M = j + 8*lh, N = lr (per 16x16 tile) ---
#pragma unroll
    for (int mi = 0; mi < 2; ++mi) {
#pragma unroll
        for (int ni = 0; ni < 2; ++ni) {
            const int    col     = oBase + waveN * 32 + ni * 16 + lr;
            const size_t rowBase = (size_t)(bBase + waveM * 32 + mi * 16 + lh * 8);
#pragma unroll
            for (int j = 0; j < 8; ++j)
                y[((rowBase + j) * GDIM + g) * DOUT + col] = acc[mi][ni][j];
        }
    }
}

extern "C" void kernel_launch(void* const* d_in, const int* in_sizes, int n_in,
                              void* d_out, int out_size, void* d_ws, size_t ws_size,
                              hipStream_t stream) {
    const float* x  = (const float*)d_in[0];   // [B, G, DIN]
    const float* W  = (const float*)d_in[1];   // [G, DOUT, DIN]
    const float* b  = (const float*)d_in[2];   // [G, DOUT]
    float*       y  = (float*)d_out;           // [B, G, DOUT]

    dim3 grid(DOUT / BN, BDIM / BM, GDIM);     // (4, 64, 16)
    multilinear_wmma_f32<<<grid, 256, 0, stream>>>(x, W, b, y);
}